// Eng_PyGT_TGCN_78589311582430
// MI455X (gfx1250) — compile-verified
//
#include <hip/hip_runtime.h>
#include <hip/hip_bf16.h>

typedef __attribute__((ext_vector_type(2))) float v2f;
typedef __attribute__((ext_vector_type(8))) float v8f;

#define HDIM 32
#define GOUT 96     // 3*H
#define KCHUNK 32   // K-tile staged in LDS per pipeline stage

#if defined(__gfx1250__) &&                                                  \
    __has_builtin(__builtin_amdgcn_global_load_async_to_lds_b128) &&         \
    __has_builtin(__builtin_amdgcn_s_wait_asynccnt)
#define USE_ASYNC_LDS 1
#else
#define USE_ASYNC_LDS 0
#endif

#if USE_ASYNC_LDS
// builtin expects gcc-vector int4 pointers, global (AS1) src / LDS (AS3) dst
typedef int b128v __attribute__((vector_size(16)));
typedef __attribute__((address_space(1))) b128v* gptr128;
typedef __attribute__((address_space(3))) b128v* lptr128;
#endif

// 16-byte global -> LDS copy (async / ASYNCcnt path when available)
__device__ __forceinline__ void copy16_g2l(const float* g, float* l) {
#if USE_ASYNC_LDS
  __builtin_amdgcn_global_load_async_to_lds_b128((gptr128)g, (lptr128)l, 0, 0);
#else
  *(float4*)l = *(const float4*)g;
#endif
}

__device__ __forceinline__ void async_join_barrier() {
#if USE_ASYNC_LDS
  __builtin_amdgcn_s_wait_asynccnt(0);
#endif
  __syncthreads();
}

// ---------------------------------------------------------------------------
// utility kernels
// ---------------------------------------------------------------------------
__global__ void zero_kernel(float* __restrict__ p, long long n) {
  long long i = blockIdx.x * (long long)blockDim.x + threadIdx.x;
  if (i < n) p[i] = 0.0f;
}

__global__ void deg_kernel(const int* __restrict__ dst, const float* __restrict__ w,
                           float* __restrict__ deg, int E) {
  int e = blockIdx.x * blockDim.x + threadIdx.x;
  if (e < E) atomicAdd(&deg[dst[e]], w[e]);
}

// deg -> dinv in place
__global__ void dinv_kernel(float* __restrict__ deg, int Nn) {
  int i = blockIdx.x * blockDim.x + threadIdx.x;
  if (i < Nn) {
    float d = deg[i];
    deg[i] = (d > 0.0f) ? rsqrtf(d) : 0.0f;
  }
}

// ---------------------------------------------------------------------------
// One-time weight re-layout: wt[(k/2)*192 + c*2 + (k&1)] = Wcat[k][c]
// (c in [0,96): gate = c/32, col = c%32).  This makes the WMMA B operand
// (W[k][c], W[k+1][c]) a contiguous 8-byte pair -> single ds_load_b64, and
// makes every K-chunk of wt a contiguous 12 KB block for async staging.
// ---------------------------------------------------------------------------
__global__ void wprep_kernel(const float* __restrict__ Wz, const float* __restrict__ Wr,
                             const float* __restrict__ Wh, float* __restrict__ wt,
                             int F) {
  int idx = blockIdx.x * blockDim.x + threadIdx.x;
  if (idx >= F * GOUT) return;
  int k = idx / GOUT;
  int c = idx - k * GOUT;
  const float* W = (c < 32) ? Wz : (c < 64 ? Wr : Wh);
  int cc = c & 31;
  wt[(size_t)(k >> 1) * (2 * GOUT) + c * 2 + (k & 1)] = W[(size_t)k * HDIM + cc];
}

// ---------------------------------------------------------------------------
// XW = node_feat @ [Wz | Wr | Wh]   ([N,F] x [F,96]) with fp32 WMMA 16x16x4.
// 8 waves / block, one 16-row tile per wave, all 6 col-tiles per wave.
// K is pipelined in 32-wide chunks: the pre-interleaved W-chunk (contiguous
// 12 KB) and the X row-tile chunk (8 tiles x 16 rows x 32) are staged into
// double-buffered LDS with GLOBAL_LOAD_ASYNC_TO_LDS_B128; WMMA operands are
// then single ds_load_b64 each.
// ---------------------------------------------------------------------------
__global__ void __launch_bounds__(256)
xw_kernel(const float* __restrict__ X, const float* __restrict__ wt,
          float* __restrict__ XW, int Nn, int F) {
  __shared__ float wbuf[2][KCHUNK * GOUT];        // 2 x 12 KB, pair-interleaved
  __shared__ float xbuf[2][8 * 16 * KCHUNK];      // 2 x 16 KB

  const int tid   = threadIdx.x;
  const int wlid  = tid >> 5;                     // wave in block: 0..7
  const int lane  = tid & 31;
  const int half  = lane >> 4;
  const int n     = lane & 15;
  const int blockNode0 = blockIdx.x * 128;
  const int node0 = blockNode0 + wlid * 16;
  const bool active = (node0 < Nn);

  // cooperative copy of one K-chunk into LDS buffer `buf`
  // W part : 3072 floats contiguous        = 768 float4 transfers
  // X part : 8 tiles x 16 rows x 8 float4  = 1024 float4 transfers
  auto stage_chunk = [&](int chunk, int buf) {
    const float* wsrc = wt + (size_t)(chunk >> 1) * (2 * GOUT);
    for (int i = tid; i < 768 + 1024; i += 256) {
      if (i < 768) {
        copy16_g2l(wsrc + i * 4, &wbuf[buf][i * 4]);
      } else {
        int j   = i - 768;
        int wl  = j >> 7;            // /128
        int rem = j & 127;
        int r   = rem >> 3;
        int f4  = (rem & 7) * 4;
        int row = blockNode0 + wl * 16 + r;
        if (row < Nn) {
          copy16_g2l(X + (size_t)row * F + chunk + f4,
                     &xbuf[buf][wl * (16 * KCHUNK) + r * KCHUNK + f4]);
        }
      }
    }
  };

  v8f acc[6];
#pragma unroll
  for (int t = 0; t < 6; ++t)
#pragma unroll
    for (int v = 0; v < 8; ++v) acc[t][v] = 0.0f;

  const int nchunks = F / KCHUNK;
  stage_chunk(0, 0);

  for (int c = 0; c < nchunks; ++c) {
    const int buf = c & 1;
    async_join_barrier();                       // buf ready for all waves
    if (c + 1 < nchunks) stage_chunk((c + 1) * KCHUNK, buf ^ 1);

    if (active) {
      const float* xl = &xbuf[buf][wlid * (16 * KCHUNK) + n * KCHUNK];
#pragma unroll
      for (int kk = 0; kk < KCHUNK; kk += 4) {
        int kl = kk + half * 2;                 // even
        v2f a = *(const v2f*)(xl + kl);         // ds_load_b64
        const float* wrow = &wbuf[buf][(kl >> 1) * (2 * GOUT)];
#pragma unroll
        for (int t = 0; t < 6; ++t) {
          v2f b = *(const v2f*)(wrow + (t * 16 + n) * 2);   // ds_load_b64
          acc[t] = __builtin_amdgcn_wmma_f32_16x16x4_f32(
              false, a, false, b, (short)0, acc[t], false, false);
        }
      }
    }
    __syncthreads();                            // done reading buf
  }

  if (active) {
#pragma unroll
    for (int t = 0; t < 6; ++t) {
      int colOut = t * 16 + n;
#pragma unroll
      for (int v = 0; v < 8; ++v) {
        int row = node0 + v + 8 * half;
        if (row < Nn) XW[(size_t)row * GOUT + colOut] = acc[t][v];
      }
    }
  }
}

// ---------------------------------------------------------------------------
// agg[dst] += xw[src] * (dinv[src]*w*dinv[dst])  — L2-resident scatter.
// One thread per (edge, group-of-4-cols): float4 load + 4 global f32 atomics.
// ---------------------------------------------------------------------------
__global__ void __launch_bounds__(256)
agg_kernel(const int* __restrict__ src, const int* __restrict__ dst,
           const float* __restrict__ w, const float* __restrict__ dinv,
           const float* __restrict__ xw, float* __restrict__ agg, int E) {
  long long idx = blockIdx.x * (long long)blockDim.x + threadIdx.x;
  int e = (int)(idx / 24);
  int g = (int)(idx % 24);
  if (e >= E) return;
  int s = src[e], d = dst[e];
  float nrm = dinv[s] * w[e] * dinv[d];
  const float4 x4 = *(const float4*)(xw + (size_t)s * GOUT + g * 4);
  float* ap = agg + (size_t)d * GOUT + g * 4;
  atomicAdd(ap + 0, x4.x * nrm);
  atomicAdd(ap + 1, x4.y * nrm);
  atomicAdd(ap + 2, x4.z * nrm);
  atomicAdd(ap + 3, x4.w * nrm);
}

// ---------------------------------------------------------------------------
// Small fused-gate GEMM: out = [aggGate + bG | second] @ Lw + Lb
// ---------------------------------------------------------------------------
__device__ __forceinline__ void
gate_gemm(const float* __restrict__ aggGate,   // agg + node0*96 + gateOff
          const float* __restrict__ bG,
          const float* second,                 // [16][32] row-major (global or LDS)
          const float* __restrict__ Lw,        // [64][32]
          const float* __restrict__ Lb,        // [32]
          int lane, v8f& o0, v8f& o1) {
  int row  = lane & 15;
  int half = lane >> 4;
  int n    = lane & 15;
  float bias0 = Lb[n];
  float bias1 = Lb[16 + n];
  v8f c0, c1;
#pragma unroll
  for (int v = 0; v < 8; ++v) { c0[v] = bias0; c1[v] = bias1; }

#pragma unroll
  for (int kk = 0; kk < 64; kk += 4) {
    int k = kk + half * 2;
    v2f a;
    if (kk < 32) {                      // gcn-conv half of the concat
      a.x = aggGate[row * GOUT + k] + bG[k];
      a.y = aggGate[row * GOUT + k + 1] + bG[k + 1];
    } else {                            // hidden-state half
      int k2 = k - 32;
      a.x = second[row * HDIM + k2];
      a.y = second[row * HDIM + k2 + 1];
    }
    v2f b0, b1;
    b0.x = Lw[k * HDIM + n];            b0.y = Lw[(k + 1) * HDIM + n];
    b1.x = Lw[k * HDIM + 16 + n];       b1.y = Lw[(k + 1) * HDIM + 16 + n];
    c0 = __builtin_amdgcn_wmma_f32_16x16x4_f32(false, a, false, b0, (short)0, c0, false, false);
    c1 = __builtin_amdgcn_wmma_f32_16x16x4_f32(false, a, false, b1, (short)0, c1, false, false);
  }
  o0 = c0;
  o1 = c1;
}

__device__ __forceinline__ float sigmoid_f(float x) {
  return 1.0f / (1.0f + __expf(-x));
}

// ---------------------------------------------------------------------------
// Fused GRU gates + output head.  One wave (32 threads) per 16-node tile.
// ---------------------------------------------------------------------------
__global__ void __launch_bounds__(32)
gate_kernel(const float* __restrict__ agg, const float* __restrict__ hidden,
            const float* __restrict__ bz, const float* __restrict__ br,
            const float* __restrict__ bh,
            const float* __restrict__ Lzw, const float* __restrict__ Lzb,
            const float* __restrict__ Lrw, const float* __restrict__ Lrb,
            const float* __restrict__ Lhw, const float* __restrict__ Lhb,
            const float* __restrict__ hw, const float* __restrict__ hb,
            float* __restrict__ out_y, float* __restrict__ out_h, int Nn) {
  __shared__ float hsr[16 * HDIM];      // Hs * R staged for the h-gate A matrix

  int node0 = blockIdx.x * 16;
  if (node0 + 16 > Nn) return;
  int lane = threadIdx.x;
  int half = lane >> 4;
  int n    = lane & 15;

  const float* aggBase = agg + (size_t)node0 * GOUT;
  const float* hsBase  = hidden + (size_t)node0 * HDIM;   // [16][32] row-major

  v8f z0, z1, r0, r1;
  gate_gemm(aggBase + 0,  bz, hsBase, Lzw, Lzb, lane, z0, z1);
  gate_gemm(aggBase + 32, br, hsBase, Lrw, Lrb, lane, r0, r1);

  // Hs in C/D layout + sigmoid activations + stage Hs*R into LDS (A layout)
  v8f hs0, hs1;
#pragma unroll
  for (int v = 0; v < 8; ++v) {
    int row = v + 8 * half;
    hs0[v] = hidden[(size_t)(node0 + row) * HDIM + n];
    hs1[v] = hidden[(size_t)(node0 + row) * HDIM + 16 + n];
    z0[v] = sigmoid_f(z0[v]);
    z1[v] = sigmoid_f(z1[v]);
    float rr0 = sigmoid_f(r0[v]);
    float rr1 = sigmoid_f(r1[v]);
    hsr[row * HDIM + n]      = hs0[v] * rr0;
    hsr[row * HDIM + 16 + n] = hs1[v] * rr1;
  }
  __syncthreads();   // 1-wave workgroup: cheap (dscnt wait)

  v8f t0, t1;
  gate_gemm(aggBase + 64, bh, hsr, Lhw, Lhb, lane, t0, t1);

  float hbv  = hb[0];
  float hwn0 = hw[n];
  float hwn1 = hw[16 + n];
#pragma unroll
  for (int v = 0; v < 8; ++v) {
    int row = node0 + v + 8 * half;
    float ht0 = tanhf(t0[v]);
    float ht1 = tanhf(t1[v]);
    float h0 = z0[v] * hs0[v] + (1.0f - z0[v]) * ht0;
    float h1 = z1[v] * hs1[v] + (1.0f - z1[v]) * ht1;
    out_h[(size_t)row * HDIM + n]      = h0;
    out_h[(size_t)row * HDIM + 16 + n] = h1;
    // y = relu(h) @ head_w + head_b : reduce 32 cols (2 per lane x 16 lanes)
    float p = fmaxf(h0, 0.0f) * hwn0 + fmaxf(h1, 0.0f) * hwn1;
    p += __shfl_xor(p, 1, 16);
    p += __shfl_xor(p, 2, 16);
    p += __shfl_xor(p, 4, 16);
    p += __shfl_xor(p, 8, 16);
    if (n == 0) out_y[row] = p + hbv;
  }
}

// ---------------------------------------------------------------------------
extern "C" void kernel_launch(void* const* d_in, const int* in_sizes, int n_in,
                              void* d_out, int out_size, void* d_ws, size_t ws_size,
                              hipStream_t stream) {
  const float* node_feat = (const float*)d_in[0];
  const int*   eidx      = (const int*)d_in[1];
  const float* ew        = (const float*)d_in[2];
  const float* hs        = (const float*)d_in[3];
  const float* Wz = (const float*)d_in[4];  const float* bz = (const float*)d_in[5];
  const float* Wr = (const float*)d_in[6];  const float* br = (const float*)d_in[7];
  const float* Wh = (const float*)d_in[8];  const float* bh = (const float*)d_in[9];
  const float* Lzw = (const float*)d_in[10]; const float* Lzb = (const float*)d_in[11];
  const float* Lrw = (const float*)d_in[12]; const float* Lrb = (const float*)d_in[13];
  const float* Lhw = (const float*)d_in[14]; const float* Lhb = (const float*)d_in[15];
  const float* hwp = (const float*)d_in[16]; const float* hbp = (const float*)d_in[17];

  int E  = in_sizes[2];
  int Nn = in_sizes[3] / HDIM;
  int F  = in_sizes[0] / Nn;

  const int* srcp = eidx;
  const int* dstp = eidx + E;

  float* ws  = (float*)d_ws;
  float* deg = ws;                         // N floats (becomes dinv in place)
  float* xw  = ws + Nn;                    // N*96
  float* agg = xw + (size_t)Nn * GOUT;     // N*96
  float* wt  = agg + (size_t)Nn * GOUT;    // F*96 (pair-interleaved weights)

  float* out_y = (float*)d_out;            // [N]
  float* out_h = out_y + Nn;               // [N,32]

  // zero accumulators (every call: deterministic under graph replay)
  {
    long long nz = Nn;
    zero_kernel<<<(int)((nz + 255) / 256), 256, 0, stream>>>(deg, nz);
    nz = (long long)Nn * GOUT;
    zero_kernel<<<(int)((nz + 255) / 256), 256, 0, stream>>>(agg, nz);
  }

  deg_kernel<<<(E + 255) / 256, 256, 0, stream>>>(dstp, ew, deg, E);
  dinv_kernel<<<(Nn + 255) / 256, 256, 0, stream>>>(deg, Nn);

  wprep_kernel<<<(F * GOUT + 255) / 256, 256, 0, stream>>>(Wz, Wr, Wh, wt, F);

  int tilesM = (Nn + 15) / 16;
  xw_kernel<<<(tilesM + 7) / 8, 256, 0, stream>>>(node_feat, wt, xw, Nn, F);

  long long aggThreads = (long long)E * 24;
  agg_kernel<<<(int)((aggThreads + 255) / 256), 256, 0, stream>>>(
      srcp, dstp, ew, deg /* = dinv */, xw, agg, E);

  gate_kernel<<<tilesM, 32, 0, stream>>>(agg, hs, bz, br, bh,
                                         Lzw, Lzb, Lrw, Lrb, Lhw, Lhb,
                                         hwp, hbp, out_y, out_h, Nn);
}